// ConvCRF_37039797961191
// MI455X (gfx1250) — compile-verified
//
#include <hip/hip_runtime.h>

// Problem constants (from reference): B=16, C=1, H=W=512, 3x3 kernel, 10 iters, ratio 0.5
#define B_    16
#define H_    512
#define W_    512
#define HW_   (H_ * W_)          // 262144 = 2^18
#define NTOT  (B_ * HW_)         // 4194304
#define ITERS 10

typedef _Float16 half_t;

// Tile shape for the iterate kernel: 64x8 outputs per 512-thread block (16 waves of 32)
#define TILE_X 64
#define TILE_Y 8
#define SW_    (TILE_X + 2)      // 66
#define SH_    (TILE_Y + 2)      // 10

#if __has_builtin(__builtin_amdgcn_global_load_async_to_lds_b32) && \
    __has_builtin(__builtin_amdgcn_s_wait_asynccnt)
#define USE_ASYNC_LDS 1
#else
#define USE_ASYNC_LDS 0
#endif

// The gfx1250 async-load builtin takes address-space-qualified int pointers:
//   (addrspace(1) int* gsrc, addrspace(3) int* ldst, imm offset, imm cpol)
typedef __attribute__((address_space(1))) int glob_i32_t;
typedef __attribute__((address_space(3))) int lds_i32_t;

// ---------------------------------------------------------------------------
// Per-pixel 3x3 softmax weights (matches reference: OOB entries are 0 BEFORE
// softmax, so they contribute exp(0 - max) to the denominator).
// ---------------------------------------------------------------------------
__device__ __forceinline__ void compute_weights(const float* __restrict__ img,
                                                int b, int y, int x,
                                                float t0, float t1, float t2,
                                                float wout[9]) {
  const float* plane = img + (size_t)b * HW_;
  const float c = plane[y * W_ + x];
  float a[9];
  int k = 0;
#pragma unroll
  for (int dx = -1; dx <= 1; ++dx) {
#pragma unroll
    for (int dy = -1; dy <= 1; ++dy) {
      const int yy = y + dx, xx = x + dy;
      const bool valid = (yy >= 0) & (yy < H_) & (xx >= 0) & (xx < W_);
      float e = 0.0f;
      if (valid) {
        const float d = 255.0f * (plane[yy * W_ + xx] - c);
        const float s = t0 * (float)(dx * dx) + t1 * (float)(dy * dy) + t2 * d * d;
        e = __expf(-0.5f * s);
      }
      a[k++] = e;
    }
  }
  float m = a[0];
#pragma unroll
  for (int i = 1; i < 9; ++i) m = fmaxf(m, a[i]);
  float z = 0.0f;
#pragma unroll
  for (int i = 0; i < 9; ++i) { a[i] = __expf(a[i] - m); z += a[i]; }
  const float inv = 1.0f / z;
#pragma unroll
  for (int i = 0; i < 9; ++i) wout[i] = a[i] * inv;
}

// ---------------------------------------------------------------------------
// Kernel 1: build fp16 softmax-kernel planes, layout [k][b][y][x] (9 planes).
// ---------------------------------------------------------------------------
__global__ void __launch_bounds__(256) crf_build_kernel(const float* __restrict__ img,
                                                        const float* __restrict__ theta,
                                                        half_t* __restrict__ kern) {
  const int p = blockIdx.x * 256 + threadIdx.x;
  const int b = p >> 18;
  const int rem = p & (HW_ - 1);
  const int y = rem >> 9;
  const int x = rem & (W_ - 1);
  const float t0 = theta[0], t1 = theta[1], t2 = theta[2];
  float w[9];
  compute_weights(img, b, y, x, t0, t1, t2, w);
#pragma unroll
  for (int k = 0; k < 9; ++k)
    kern[(size_t)k * NTOT + p] = (half_t)w[k];
}

// ---------------------------------------------------------------------------
// Kernel 2: one mean-field iteration. Halo tile of predIn staged into LDS via
// CDNA5 async global->LDS loads (ASYNCcnt), then 9-tap weighted gather.
// ---------------------------------------------------------------------------
__global__ void __launch_bounds__(TILE_X * TILE_Y)
crf_iterate(const half_t* __restrict__ kern,
            const float* __restrict__ unary,
            const float* __restrict__ predIn,
            float* __restrict__ predOut,
            const float* __restrict__ weight) {
  __shared__ float smem[SH_ * SW_];

  const int tx = threadIdx.x, ty = threadIdx.y;
  const int bx = blockIdx.x * TILE_X;
  const int by = blockIdx.y * TILE_Y;
  const int b  = blockIdx.z;
  const float* pin = predIn + (size_t)b * HW_;

  // --- stage (TILE_Y+2) x (TILE_X+2) halo tile of predIn into LDS ---
  const int flat = ty * TILE_X + tx;
  for (int i = flat; i < SH_ * SW_; i += TILE_X * TILE_Y) {
    const int r = i / SW_;
    const int c = i - r * SW_;
    const int gy = by + r - 1;
    const int gx = bx + c - 1;
    const bool valid = (gy >= 0) & (gy < H_) & (gx >= 0) & (gx < W_);
#if USE_ASYNC_LDS
    if (valid) {
      __builtin_amdgcn_global_load_async_to_lds_b32(
          (glob_i32_t*)(pin + gy * W_ + gx),
          (lds_i32_t*)(&smem[i]),
          /*offset=*/0, /*cpol=*/0);
    } else {
      smem[i] = 0.0f;  // zero-padded boundary
    }
#else
    smem[i] = valid ? pin[gy * W_ + gx] : 0.0f;
#endif
  }
#if USE_ASYNC_LDS
  __builtin_amdgcn_s_wait_asynccnt(0);
#endif
  __syncthreads();

  // --- 9-tap spatially-varying gather ---
  const int y = by + ty;
  const int x = bx + tx;
  const size_t p = (size_t)b * HW_ + (size_t)y * W_ + x;
  const half_t* kp = kern + p;

  float msg = 0.0f;
#pragma unroll
  for (int k = 0; k < 9; ++k) {
    const int dx = k / 3 - 1;   // row offset
    const int dy = k % 3 - 1;   // col offset
    const float kw = (float)kp[(size_t)k * NTOT];
    const float nb = smem[(ty + 1 + dx) * SW_ + (tx + 1 + dy)];
    msg = fmaf(kw, nb, msg);
  }
  const float w = weight[0];
  predOut[p] = fmaf(0.5f * w, msg, 0.5f * unary[p]);
}

// ---------------------------------------------------------------------------
// Fallback (small workspace): fused rebuild-weights + message pass per iter.
// ---------------------------------------------------------------------------
__global__ void __launch_bounds__(256)
crf_fused_iter(const float* __restrict__ img,
               const float* __restrict__ theta,
               const float* __restrict__ weight,
               const float* __restrict__ unary,
               const float* __restrict__ predIn,
               float* __restrict__ predOut) {
  const int p = blockIdx.x * 256 + threadIdx.x;
  const int b = p >> 18;
  const int rem = p & (HW_ - 1);
  const int y = rem >> 9;
  const int x = rem & (W_ - 1);

  float wk[9];
  compute_weights(img, b, y, x, theta[0], theta[1], theta[2], wk);

  const float* pin = predIn + (size_t)b * HW_;
  float msg = 0.0f;
  int k = 0;
#pragma unroll
  for (int dx = -1; dx <= 1; ++dx) {
#pragma unroll
    for (int dy = -1; dy <= 1; ++dy) {
      const int yy = y + dx, xx = x + dy;
      const bool valid = (yy >= 0) & (yy < H_) & (xx >= 0) & (xx < W_);
      const float nb = valid ? pin[yy * W_ + xx] : 0.0f;
      msg = fmaf(wk[k++], nb, msg);
    }
  }
  predOut[p] = fmaf(0.5f * weight[0], msg, 0.5f * unary[p]);
}

// ---------------------------------------------------------------------------
extern "C" void kernel_launch(void* const* d_in, const int* in_sizes, int n_in,
                              void* d_out, int out_size, void* d_ws, size_t ws_size,
                              hipStream_t stream) {
  const float* image  = (const float*)d_in[0];  // (16,1,512,512)
  const float* unary  = (const float*)d_in[1];  // (16,1,512,512)
  const float* theta  = (const float*)d_in[2];  // (1,3,1,1)
  const float* weight = (const float*)d_in[3];  // (1,)
  float* out = (float*)d_out;                   // (16,1,512,512)

  const size_t kernBytes = (size_t)9 * NTOT * sizeof(half_t);  // ~75.5 MB
  const size_t predBytes = (size_t)NTOT * sizeof(float);       // ~16.8 MB

  if (ws_size >= kernBytes + 2 * predBytes) {
    // Fast path: precompute fp16 softmax kernel once; everything L2-resident.
    half_t* kern = (half_t*)d_ws;
    float*  p0   = (float*)((char*)d_ws + kernBytes);
    float*  p1   = (float*)((char*)d_ws + kernBytes + predBytes);

    crf_build_kernel<<<NTOT / 256, 256, 0, stream>>>(image, theta, kern);

    dim3 blk(TILE_X, TILE_Y, 1);
    dim3 grd(W_ / TILE_X, H_ / TILE_Y, B_);
    const float* in = unary;
    for (int it = 0; it < ITERS; ++it) {
      float* o = (it == ITERS - 1) ? out : ((it & 1) ? p1 : p0);
      crf_iterate<<<grd, blk, 0, stream>>>(kern, unary, in, o, weight);
      in = o;
    }
  } else if (ws_size >= predBytes) {
    // Fallback: recompute weights every iteration; ping-pong ws <-> out.
    float* p0 = (float*)d_ws;
    const float* in = unary;
    for (int it = 0; it < ITERS; ++it) {
      float* o = (it & 1) ? out : p0;   // it==9 (odd) -> out
      crf_fused_iter<<<NTOT / 256, 256, 0, stream>>>(image, theta, weight,
                                                     unary, in, o);
      in = o;
    }
  }
  // (ws_size < 16.8 MB is not expected for this problem size.)
}